// GCNBlock_6820408066437
// MI455X (gfx1250) — compile-verified
//
#include <hip/hip_runtime.h>
#include <stdint.h>

// Problem sizes (fixed by the reference)
#define GN 4096      // nodes
#define GD 16        // feature dim
#define GB 64        // batch
#define NEG_SLOPE 0.2f

// Stage-B tiling
#define KC     128              // K-chunk staged in LDS
#define KROW   (KC + 4)         // 132: row stride. 528B row pitch is 16B-aligned (async b128 OK);
                                // 132 mod 64 = 4 -> lanes hit banks {4n,4n+1}: conflict-free b64 reads
#define RT     2                // row-tiles (16 adj rows each) per workgroup
#define BBATCH 4                // batches per workgroup (RT*BBATCH = 8 waves = 256 threads)
#define AFLT   (RT * 16 * KROW)      // 4224 floats: adj tile
#define BFLT   (BBATCH * 16 * KROW)  // 8448 floats: hT tile
#define BUFF   (AFLT + BFLT)         // 12672 floats per buffer (double-buffered: ~99 KB LDS)

typedef __attribute__((ext_vector_type(2))) float v2f;
typedef __attribute__((ext_vector_type(8))) float v8f;

// Async global->LDS copy (CDNA5 GLOBAL_LOAD_ASYNC_TO_LDS_B128, tracked by ASYNCcnt).
// VDST operand = wave-relative LDS byte offset = low 32 bits of the generic pointer
// (as3->flat cast keeps the LDS offset in addr[31:0], ISA §10.2).  Deriving the
// operand from the real LDS pointer also captures the array, so the compiler must
// treat these asm statements as writers of `lds` (fixes round-2 dead-LDS folding).
__device__ __forceinline__ void async_ld_b128(const float* g, const float* l) {
    asm volatile("global_load_async_to_lds_b128 %0, %1, off"
                 :: "v"((uint32_t)(uintptr_t)l), "v"(g)
                 : "memory");
}

__device__ __forceinline__ void wait_async0() {
#if __has_builtin(__builtin_amdgcn_s_wait_asynccnt)
    __builtin_amdgcn_s_wait_asynccnt(0);
#else
    asm volatile("s_wait_asynccnt 0x0" ::: "memory");
#endif
}

// ---------------------------------------------------------------------------
// Stage A: hT[b][d][n] = leaky_relu( sum_e x[b][n][e] * W[e][d] )
// Memory bound; transposed write makes Stage-B B-operand reads contiguous.
// ---------------------------------------------------------------------------
__global__ __launch_bounds__(256) void gcn_linear_lrelu_T(
    const float* __restrict__ xin, const float* __restrict__ W,
    float* __restrict__ hT)
{
    __shared__ float sW[GD * GD];
    const int tid = threadIdx.x;
    sW[tid] = W[tid];
    __syncthreads();

    const int t = blockIdx.x * 256 + tid;   // t = b*GN + n
    const int b = t >> 12;
    const int n = t & (GN - 1);

    const float* xr = xin + (size_t)t * GD;
    float xv[GD];
#pragma unroll
    for (int e = 0; e < GD; e += 4) {
        float4 v = *(const float4*)(xr + e);
        xv[e + 0] = v.x; xv[e + 1] = v.y; xv[e + 2] = v.z; xv[e + 3] = v.w;
    }
#pragma unroll
    for (int d = 0; d < GD; ++d) {
        float acc = 0.0f;
#pragma unroll
        for (int e = 0; e < GD; ++e) acc += xv[e] * sW[e * GD + d];
        acc = (acc > 0.0f) ? acc : NEG_SLOPE * acc;
        hT[(size_t)(b * GD + d) * GN + n] = acc;
    }
}

// ---------------------------------------------------------------------------
// Stage B: out[b][n][d] = sum_k adj[n][k] * h[b][k][d] + h[b][n][d]
// fp32 WMMA 16x16x4. 8 waves = RT row-tiles x BBATCH batches.
// Double-buffered LDS filled by async global->LDS copies: chunk k+1 streams in
// while WMMAs consume chunk k. One barrier per chunk.
// ---------------------------------------------------------------------------
__global__ __launch_bounds__(256) void gcn_adj_wmma(
    const float* __restrict__ adj, const float* __restrict__ hT,
    float* __restrict__ out)
{
    __shared__ __align__(16) float lds[2 * BUFF];

    const int tid  = threadIdx.x;
    const int lane = tid & 31;
    const int wave = tid >> 5;
    const int rt   = wave & (RT - 1);
    const int bi   = wave >> 1;
    const int nl   = lane & 15;
    const int half = lane >> 4;         // ISA: lanes 16-31 handle K+2 (A) / M+8 (C/D)

    const int rowbase = blockIdx.x * (RT * 16);
    const int bbase   = blockIdx.y * BBATCH;
    const int b       = bbase + bi;
    const int n0      = rowbase + rt * 16;

    // Per-thread async-copy descriptors: 4x float4 of adj + 8x float4 of hT per chunk.
    // pA/pB point into buffer 0; add (s^1)*BUFF at issue time (pointer arithmetic
    // keeps LDS provenance visible to the compiler).
    const float* gA[4]; float* pA[4];
#pragma unroll
    for (int i = 0; i < 4; ++i) {
        int q = tid + i * 256, row = q >> 5, c4 = q & 31;   // 32 rows x 32 float4
        gA[i] = adj + (size_t)(rowbase + row) * GN + c4 * 4;
        pA[i] = &lds[row * KROW + c4 * 4];
    }
    const float* gB[8]; float* pB[8];
#pragma unroll
    for (int i = 0; i < 8; ++i) {
        int q = tid + i * 256, row = q >> 5, c4 = q & 31;   // 64 rows x 32 float4
        gB[i] = hT + (size_t)((bbase + (row >> 4)) * GD + (row & 15)) * GN + c4 * 4;
        pB[i] = &lds[AFLT + row * KROW + c4 * 4];
    }

    // Seed accumulator with identity term: C[M][N] = h[b][n0+M][N] = hT[b][N][n0+M]
    v8f c;
    const float* hTcol = hT + (size_t)(b * GD + nl) * GN + n0 + half * 8;
#pragma unroll
    for (int r = 0; r < 8; ++r) c[r] = hTcol[r];

    // Prologue: stream chunk 0 into buffer 0.
#pragma unroll
    for (int i = 0; i < 4; ++i) async_ld_b128(gA[i], pA[i]);
#pragma unroll
    for (int i = 0; i < 8; ++i) async_ld_b128(gB[i], pB[i]);

    int s = 0;
    for (int k0 = 0; k0 < GN; k0 += KC, s ^= 1) {
        wait_async0();          // my chunk-k copies into buffer s are done
        __syncthreads();        // everyone's copies done AND everyone done reading buf s^1

        if (k0 + KC < GN) {     // stream chunk k+1 into the other buffer
            const int soff = (s ^ 1) * BUFF;
#pragma unroll
            for (int i = 0; i < 4; ++i) async_ld_b128(gA[i] + k0 + KC, pA[i] + soff);
#pragma unroll
            for (int i = 0; i < 8; ++i) async_ld_b128(gB[i] + k0 + KC, pB[i] + soff);
        }

        // A lane layout (ISA 16x4 f32): lane<16 -> K=kk,kk+1 ; lane>=16 -> K=kk+2,kk+3
        const float* aRow = &lds[s * BUFF + (rt * 16 + nl) * KROW + half * 2];
        const float* bRow = &lds[s * BUFF + AFLT + (bi * 16 + nl) * KROW + half * 2];
#pragma unroll 8
        for (int kk = 0; kk < KC; kk += 4) {
            v2f a  = *(const v2f*)(aRow + kk);
            v2f bb = *(const v2f*)(bRow + kk);
            c = __builtin_amdgcn_wmma_f32_16x16x4_f32(
                    false, a, false, bb, (short)0, c, false, false);
        }
    }

    // Store D: element (M,N) -> out[b][n0+M][N];  M = r + half*8, N = nl
    float* outb = out + (size_t)b * GN * GD;
#pragma unroll
    for (int r = 0; r < 8; ++r) {
        outb[(size_t)(n0 + half * 8 + r) * GD + nl] = c[r];
    }
}

// ---------------------------------------------------------------------------
extern "C" void kernel_launch(void* const* d_in, const int* in_sizes, int n_in,
                              void* d_out, int out_size, void* d_ws, size_t ws_size,
                              hipStream_t stream) {
    (void)in_sizes; (void)n_in; (void)out_size; (void)ws_size;
    const float* x   = (const float*)d_in[0];
    const float* adj = (const float*)d_in[1];
    // d_in[2] = Identity: unused (I @ h == h, added exactly via accumulator seed)
    const float* W[3] = { (const float*)d_in[3], (const float*)d_in[4], (const float*)d_in[5] };
    float* out = (float*)d_out;
    float* hT  = (float*)d_ws;                // 64*16*4096*4 = 16 MB scratch

    dim3 blk(256);
    dim3 grdA((GB * GN) / 256);               // 1024 blocks
    dim3 grdB(GN / (RT * 16), GB / BBATCH);   // 128 x 16 blocks

    const float* cur = x;
    for (int l = 0; l < 3; ++l) {
        gcn_linear_lrelu_T<<<grdA, blk, 0, stream>>>(cur, W[l], hT);
        gcn_adj_wmma<<<grdB, blk, 0, stream>>>(adj, hT, out);
        cur = out;   // safe: stage B reads only adj + hT
    }
}